// PointNetFeaturePropagation5_53154515256140
// MI455X (gfx1250) — compile-verified
//
#include <hip/hip_runtime.h>
#include <hip/hip_bf16.h>

typedef float v2f __attribute__((ext_vector_type(2)));
typedef float v8f __attribute__((ext_vector_type(8)));

#define B_   4
#define N_   8192
#define S_   2048
#define C1_  128
#define C2_  256
#define CIN0 (C1_ + C2_)   // 384
#define CO0  256
#define CO1  128
#define BN_  (B_ * N_)     // 32768

// ---------------- Kernel 1: 3-NN search -> indices + normalized weights ----
// wknn layout (32-bit planes of BN_): [i0][i1][i2][w0][w1][w2]
__global__ __launch_bounds__(256) void knn_kernel(
    const float* __restrict__ xyz1, const float* __restrict__ xyz2,
    float* __restrict__ wknn)
{
    __shared__ __align__(16) float x2[S_], y2[S_], z2[S_];
    const int b = blockIdx.y;
    const int tid = threadIdx.x;
    for (int i = tid; i < S_; i += 256) {
        x2[i] = xyz2[(b * 3 + 0) * S_ + i];
        y2[i] = xyz2[(b * 3 + 1) * S_ + i];
        z2[i] = xyz2[(b * 3 + 2) * S_ + i];
    }
    __syncthreads();

    const int n = blockIdx.x * 256 + tid;
    const float px = xyz1[(b * 3 + 0) * N_ + n];
    const float py = xyz1[(b * 3 + 1) * N_ + n];
    const float pz = xyz1[(b * 3 + 2) * N_ + n];

    float d0 = 3.4e38f, d1 = 3.4e38f, d2 = 3.4e38f;
    int   i0 = 0,       i1 = 0,       i2 = 0;
    for (int s0 = 0; s0 < S_; s0 += 4) {
        const float4 xs = *(const float4*)&x2[s0];
        const float4 ys = *(const float4*)&y2[s0];
        const float4 zs = *(const float4*)&z2[s0];
        float dd[4];
        {
            float dx = px - xs.x, dy = py - ys.x, dz = pz - zs.x;
            dd[0] = fmaf(dx, dx, fmaf(dy, dy, dz * dz));
            dx = px - xs.y; dy = py - ys.y; dz = pz - zs.y;
            dd[1] = fmaf(dx, dx, fmaf(dy, dy, dz * dz));
            dx = px - xs.z; dy = py - ys.z; dz = pz - zs.z;
            dd[2] = fmaf(dx, dx, fmaf(dy, dy, dz * dz));
            dx = px - xs.w; dy = py - ys.w; dz = pz - zs.w;
            dd[3] = fmaf(dx, dx, fmaf(dy, dy, dz * dz));
        }
#pragma unroll
        for (int j = 0; j < 4; ++j) {
            const float d = dd[j];
            const int   s = s0 + j;
            if (d < d2) {
                if (d < d1) {
                    d2 = d1; i2 = i1;
                    if (d < d0) { d1 = d0; i1 = i0; d0 = d; i0 = s; }
                    else        { d1 = d;  i1 = s; }
                } else { d2 = d; i2 = s; }
            }
        }
    }
    float r0 = 1.f / (d0 + 1e-8f);
    float r1 = 1.f / (d1 + 1e-8f);
    float r2 = 1.f / (d2 + 1e-8f);
    float rs = 1.f / (r0 + r1 + r2);

    const int bn = b * N_ + n;
    int* ip = (int*)wknn;
    ip[0 * BN_ + bn] = i0;
    ip[1 * BN_ + bn] = i1;
    ip[2 * BN_ + bn] = i2;
    wknn[3 * BN_ + bn] = r0 * rs;
    wknn[4 * BN_ + bn] = r1 * rs;
    wknn[5 * BN_ + bn] = r2 * rs;
}

// ---------------- WMMA pointwise-conv GEMM:  Y = W @ X + bias --------------
// Full-O tile (no output-row blocking -> X read exactly once).
// MODE 0: X = concat(points1, on-the-fly 3-NN interp of points2)
// MODE 1: X = relu(scale*Y0 + shift)   (BN0+ReLU fused on load)
// Epilogue: per-(channel, block) partial sum / sum-of-squares for BN stats,
// reduced deterministically with a fixed shfl_xor tree.
template <int CIN, int OTILE, int MODE>
__global__ __launch_bounds__(256) void gemm_wmma_kernel(
    const float* __restrict__ W,     // (OTILE, CIN)
    const float* __restrict__ bias,  // (OTILE)
    const float* __restrict__ A0,    // points1 (MODE 0) or Y0 (MODE 1)
    const float* __restrict__ P2,    // points2 (MODE 0)
    const float* __restrict__ wknn,  // knn planes (MODE 0)
    const float* __restrict__ scale, // (CIN) MODE 1
    const float* __restrict__ shift, // (CIN) MODE 1
    float* __restrict__ Y,           // (B, OTILE, N)
    float* __restrict__ partials)    // (2, OTILE, PTOT)
{
    constexpr int KC      = 32;
    constexpr int LDW     = 34;                  // pad: conflict-free, 8B aligned
    constexpr int WAVES_O = OTILE / 32;
    constexpr int WAVES_N = 8 / WAVES_O;
    constexpr int NTILE   = WAVES_N * 32;
    constexpr int PTOT    = B_ * (N_ / NTILE) * WAVES_N;   // == 1024 both layers
    static_assert(PTOT == 1024, "partial count");

    __shared__ float Wt[OTILE * LDW];            // [o][k]
    __shared__ float Xt[NTILE * LDW];            // [n][k] (transposed K-pairs)
    __shared__ int   nidx[3][NTILE];
    __shared__ float nwgt[3][NTILE];

    const int b    = blockIdx.z;
    const int nblk = blockIdx.x;                 // * NTILE columns
    const int tid  = threadIdx.x;
    const int lane = tid & 31;
    const int wave = tid >> 5;
    const int wo   = (wave % WAVES_O) * 32;      // wave's O offset
    const int wn   = (wave / WAVES_O) * 32;      // wave's N offset
    const int lh   = lane >> 4;                  // lane half
    const int l16  = lane & 15;

    if constexpr (MODE == 0) {
        if (tid < NTILE) {
            const int bn = b * N_ + nblk * NTILE + tid;
            const int* ip = (const int*)wknn;
            nidx[0][tid] = ip[0 * BN_ + bn];
            nidx[1][tid] = ip[1 * BN_ + bn];
            nidx[2][tid] = ip[2 * BN_ + bn];
            nwgt[0][tid] = wknn[3 * BN_ + bn];
            nwgt[1][tid] = wknn[4 * BN_ + bn];
            nwgt[2][tid] = wknn[5 * BN_ + bn];
        }
    }

    v8f acc[2][2];
#pragma unroll
    for (int ta = 0; ta < 2; ++ta)
#pragma unroll
        for (int v = 0; v < 8; ++v) {
            float bz = bias[wo + ta * 16 + v + 8 * lh];
            acc[ta][0][v] = bz;
            acc[ta][1][v] = bz;
        }

    for (int k0 = 0; k0 < CIN; k0 += KC) {
        // stage W tile: OTILE x KC
#pragma unroll
        for (int i = tid; i < OTILE * KC; i += 256) {
            int o = i >> 5, k = i & 31;
            Wt[o * LDW + k] = W[(size_t)o * CIN + k0 + k];
        }
        // stage X tile: KC x NTILE -> Xt[n][k]
#pragma unroll
        for (int i = tid; i < KC * NTILE; i += 256) {
            int k = i / NTILE, nn = i % NTILE;
            int c = k0 + k;
            int gn = nblk * NTILE + nn;
            float x;
            if constexpr (MODE == 0) {
                if (c < C1_) {
                    x = A0[((size_t)b * C1_ + c) * N_ + gn];
                } else {
                    const float* row = P2 + ((size_t)b * C2_ + (c - C1_)) * S_;
                    x = nwgt[0][nn] * row[nidx[0][nn]];
                    x = fmaf(nwgt[1][nn], row[nidx[1][nn]], x);
                    x = fmaf(nwgt[2][nn], row[nidx[2][nn]], x);
                }
            } else {
                float y = A0[((size_t)b * CIN + c) * N_ + gn];
                x = fmaxf(0.f, fmaf(scale[c], y, shift[c]));
            }
            Xt[nn * LDW + k] = x;
        }
        __syncthreads();

#pragma unroll
        for (int kk = 0; kk < KC; kk += 4) {
            const int kl = kk + 2 * lh;          // 16x4 f32 layout: K = v + 2*(lane/16)
            v2f a[2], bb[2];
#pragma unroll
            for (int ta = 0; ta < 2; ++ta) {
                const float* p = &Wt[(wo + ta * 16 + l16) * LDW + kl];
                a[ta][0] = p[0]; a[ta][1] = p[1];
            }
#pragma unroll
            for (int tb = 0; tb < 2; ++tb) {
                const float* p = &Xt[(wn + tb * 16 + l16) * LDW + kl];
                bb[tb][0] = p[0]; bb[tb][1] = p[1];
            }
#pragma unroll
            for (int ta = 0; ta < 2; ++ta)
#pragma unroll
                for (int tb = 0; tb < 2; ++tb)
                    acc[ta][tb] = __builtin_amdgcn_wmma_f32_16x16x4_f32(
                        false, a[ta], false, bb[tb],
                        (short)0, acc[ta][tb], false, false);
        }
        __syncthreads();
    }

    // ---- write result tile ----
#pragma unroll
    for (int ta = 0; ta < 2; ++ta)
#pragma unroll
        for (int tb = 0; tb < 2; ++tb)
#pragma unroll
            for (int v = 0; v < 8; ++v) {
                int o  = wo + ta * 16 + v + 8 * lh;
                int gn = nblk * NTILE + wn + tb * 16 + l16;
                Y[((size_t)b * OTILE + o) * N_ + gn] = acc[ta][tb][v];
            }

    // ---- BN partial stats (deterministic shfl tree within 16-lane groups) --
    const int p = (b * (N_ / NTILE) + nblk) * WAVES_N + (wave / WAVES_O);
#pragma unroll
    for (int ta = 0; ta < 2; ++ta)
#pragma unroll
        for (int v = 0; v < 8; ++v) {
            float s = 0.f, q = 0.f;
#pragma unroll
            for (int tb = 0; tb < 2; ++tb) {
                float t = acc[ta][tb][v];
                s += t;
                q = fmaf(t, t, q);
            }
#pragma unroll
            for (int m = 1; m < 16; m <<= 1) {
                s += __shfl_xor(s, m);
                q += __shfl_xor(q, m);
            }
            if (l16 == 0) {
                int ch = wo + ta * 16 + v + 8 * lh;
                partials[(size_t)ch * PTOT + p]              = s;
                partials[(size_t)(OTILE + ch) * PTOT + p]    = q;
            }
        }
}

// ---------------- Finalize BN stats from 1024 partials per channel ---------
template <int CO>
__global__ __launch_bounds__(256) void bn_finalize_kernel(
    const float* __restrict__ partials, const float* __restrict__ gamma,
    const float* __restrict__ beta, float* __restrict__ scale,
    float* __restrict__ shift)
{
    constexpr int P = 1024;
    __shared__ float ss[256], sq[256];
    const int c = blockIdx.x;
    const int tid = threadIdx.x;
    float s = 0.f, q = 0.f;
    for (int i = tid; i < P; i += 256) {
        s += partials[(size_t)c * P + i];
        q += partials[(size_t)(CO + c) * P + i];
    }
    ss[tid] = s; sq[tid] = q;
    __syncthreads();
    for (int w = 128; w > 0; w >>= 1) {
        if (tid < w) { ss[tid] += ss[tid + w]; sq[tid] += sq[tid + w]; }
        __syncthreads();
    }
    if (tid == 0) {
        const float M = (float)BN_;
        float mean = ss[0] / M;
        float var  = sq[0] / M - mean * mean;   // biased, like jnp.var
        float sc   = gamma[c] * rsqrtf(var + 1e-5f);
        scale[c] = sc;
        shift[c] = fmaf(-mean, sc, beta[c]);
    }
}

// ---------------- In-place BN+ReLU on final output -------------------------
__global__ __launch_bounds__(256) void bn_apply_kernel(
    float* __restrict__ Y, const float* __restrict__ scale,
    const float* __restrict__ shift)
{
    size_t idx = (size_t)blockIdx.x * 256 + threadIdx.x;
    int c = (int)((idx / N_) % CO1);
    Y[idx] = fmaxf(0.f, fmaf(scale[c], Y[idx], shift[c]));
}

extern "C" void kernel_launch(void* const* d_in, const int* in_sizes, int n_in,
                              void* d_out, int out_size, void* d_ws, size_t ws_size,
                              hipStream_t stream) {
    const float* xyz1    = (const float*)d_in[0];
    const float* xyz2    = (const float*)d_in[1];
    const float* points1 = (const float*)d_in[2];
    const float* points2 = (const float*)d_in[3];
    const float* w0      = (const float*)d_in[4];
    const float* b0      = (const float*)d_in[5];
    const float* g0      = (const float*)d_in[6];
    const float* be0     = (const float*)d_in[7];
    const float* w1      = (const float*)d_in[8];
    const float* b1      = (const float*)d_in[9];
    const float* g1      = (const float*)d_in[10];
    const float* be1     = (const float*)d_in[11];
    float* out = (float*)d_out;

    float* ws        = (float*)d_ws;
    float* wknn      = ws;                                  //  6*BN_   = 196,608 f
    float* Y0        = wknn + (size_t)6 * BN_;              //  B*CO0*N = 8,388,608 f
    float* partials0 = Y0 + (size_t)B_ * CO0 * N_;          //  2*256*1024 = 524,288 f
    float* partials1 = partials0 + (size_t)2 * CO0 * 1024;  //  2*128*1024 = 262,144 f
    float* scale0    = partials1 + (size_t)2 * CO1 * 1024;  //  256
    float* shift0    = scale0 + CO0;
    float* scale1    = shift0 + CO0;
    float* shift1    = scale1 + CO1;

    // 1) 3-NN indices + weights
    knn_kernel<<<dim3(N_ / 256, B_), 256, 0, stream>>>(xyz1, xyz2, wknn);

    // 2) GEMM0 (full-O 256x32 tile): Y0 = w0 @ concat(points1, interp) + b0
    //    + BN0 partial stats in epilogue
    gemm_wmma_kernel<CIN0, CO0, 0><<<dim3(N_ / 32, 1, B_), 256, 0, stream>>>(
        w0, b0, points1, points2, wknn, nullptr, nullptr, Y0, partials0);

    // 3) finalize BN0 -> scale0/shift0
    bn_finalize_kernel<CO0><<<CO0, 256, 0, stream>>>(partials0, g0, be0,
                                                     scale0, shift0);

    // 4) GEMM1 (128x64 tile, BN0+ReLU fused on load): out_pre = w1 @ h + b1
    //    + BN1 partial stats in epilogue
    gemm_wmma_kernel<CO0, CO1, 1><<<dim3(N_ / 64, 1, B_), 256, 0, stream>>>(
        w1, b1, Y0, nullptr, nullptr, scale0, shift0, out, partials1);

    // 5) finalize BN1 -> scale1/shift1
    bn_finalize_kernel<CO1><<<CO1, 256, 0, stream>>>(partials1, g1, be1,
                                                     scale1, shift1);

    // 6) BN1+ReLU in place on d_out
    bn_apply_kernel<<<(B_ * CO1 * N_) / 256, 256, 0, stream>>>(out, scale1, shift1);
}